// ModelNew_68161130987791
// MI455X (gfx1250) — compile-verified
//
#include <hip/hip_runtime.h>

typedef __attribute__((ext_vector_type(2))) float v2f;
typedef __attribute__((ext_vector_type(8))) float v8f;

#define BATCH 16384
#define INP   4096

// ---- stage-2 tiling -------------------------------------------------------
#define KCHUNKS 4
#define KLEN    (INP / KCHUNKS)    // 1024 K per block
#define KC      64                 // K staged per LDS chunk
#define NC      (KLEN / KC)        // 16 chunks
#define ROWS_PB 64                 // rows per block (4 waves x 16)
#define LSTRIDE (KC + 4)           // 68: bank = (4r + c) % 64 -> conflict-free
#define THREADS 128

// ---------------------------------------------------------------------------
// Kernel 0: zero the output and the wsum scratch (harness poisons them).
// ---------------------------------------------------------------------------
__global__ void mn_zero_kernel(float* __restrict__ out, float* __restrict__ wsum) {
    int i = blockIdx.x * blockDim.x + threadIdx.x;
    if (i < BATCH) out[i] = 0.0f;
    if (i < INP)   wsum[i] = 0.0f;
}

// ---------------------------------------------------------------------------
// Kernel 1: wsum[j] = sum_i W[i][j]  (column sums, pure bandwidth).
// float4 loads fully coalesced; 64 row-chunks combined with f32 atomics.
// ---------------------------------------------------------------------------
__global__ void mn_colsum_kernel(const float* __restrict__ w,
                                 float* __restrict__ wsum) {
    const int gid   = blockIdx.x * blockDim.x + threadIdx.x;   // 0..65535
    const int cg    = gid & 1023;        // column group: 4 consecutive cols
    const int chunk = gid >> 10;         // row chunk: 64 rows each, 64 chunks
    const int col   = cg << 2;
    const int row0  = chunk << 6;

    float ax = 0.0f, ay = 0.0f, az = 0.0f, aw = 0.0f;
    const float* p = w + (size_t)row0 * INP + col;
#pragma unroll 4
    for (int r = 0; r < 64; ++r) {
        float4 v = *(const float4*)p;
        ax += v.x; ay += v.y; az += v.z; aw += v.w;
        p += INP;
    }
    atomicAdd(&wsum[col + 0], ax);
    atomicAdd(&wsum[col + 1], ay);
    atomicAdd(&wsum[col + 2], az);
    atomicAdd(&wsum[col + 3], aw);
}

// ---------------------------------------------------------------------------
// Kernel 2: out[b] = 2 * dot(X[b,:], wsum) via V_WMMA_F32_16X16X4_F32,
// with X streamed through LDS using GLOBAL_LOAD_ASYNC_TO_LDS_B128
// (ASYNCcnt-tracked, double-buffered) and conflict-free ds_load_b64
// fragment reads (LDS row stride 68 floats: bank = (4r + c) % 64).
// ---------------------------------------------------------------------------
__global__ __launch_bounds__(THREADS)
void mn_gemv_wmma_kernel(const float* __restrict__ x,
                         const float* __restrict__ wsum,
                         float* __restrict__ out) {
    __shared__ float tile[2][ROWS_PB * LSTRIDE];
    __shared__ float wlds[KLEN];

    const int tid    = threadIdx.x;
    const int lane   = tid & 31;
    const int wv     = tid >> 5;              // wave in block: 0..3
    const int rowgrp = blockIdx.x >> 2;       // 0..255
    const int kchunk = blockIdx.x & 3;        // 0..3
    const int row0   = rowgrp * ROWS_PB;
    const int k0     = kchunk * KLEN;

    // Stage the wsum chunk for this block (visible after first barrier).
    for (int j = tid; j < KLEN; j += THREADS) wlds[j] = wsum[k0 + j];

    const unsigned tile0 = (unsigned)(uintptr_t)(&tile[0][0]);
    const unsigned tile1 = (unsigned)(uintptr_t)(&tile[1][0]);

    // Issue async staging of chunk 0 into buffer 0.
    // Thread handles 8 float4 groups: g = tid + j*128, row = g>>4, cg = g&15.
    {
        const int k = k0;
#pragma unroll
        for (int j = 0; j < 8; ++j) {
            const int g  = tid + j * THREADS;
            const int r  = g >> 4;
            const int cg = g & 15;
            const int      voff  = (((row0 + r) << 12) + k + (cg << 2)) << 2;
            const unsigned laddr = tile0 + (unsigned)((r * LSTRIDE + (cg << 2)) << 2);
            asm volatile("global_load_async_to_lds_b128 %0, %1, %2"
                         :: "v"(laddr), "v"(voff), "s"(x) : "memory");
        }
    }

    const int m   = lane & 15;                // A row within 16-row tile
    const int khi = (lane >> 4) << 1;         // K split: 0 (lanes 0-15) / 2
    v8f acc = {};

    for (int c = 0; c < NC; ++c) {
        if (c + 1 < NC) {
            // Prefetch next chunk into the other buffer.
            const int      k     = k0 + (c + 1) * KC;
            const unsigned tbase = ((c + 1) & 1) ? tile1 : tile0;
#pragma unroll
            for (int j = 0; j < 8; ++j) {
                const int g  = tid + j * THREADS;
                const int r  = g >> 4;
                const int cg = g & 15;
                const int      voff  = (((row0 + r) << 12) + k + (cg << 2)) << 2;
                const unsigned laddr = tbase + (unsigned)((r * LSTRIDE + (cg << 2)) << 2);
                asm volatile("global_load_async_to_lds_b128 %0, %1, %2"
                             :: "v"(laddr), "v"(voff), "s"(x) : "memory");
            }
            // 8 outstanding from chunk c+1; in-order completion means
            // asynccnt <= 8 guarantees chunk c has fully landed in LDS.
            asm volatile("s_wait_asynccnt 0x8" ::: "memory");
        } else {
            asm volatile("s_wait_asynccnt 0x0" ::: "memory");
        }
        __syncthreads();   // staged tile (and wlds on c==0) visible to all

        const float* tb     = &tile[c & 1][0];
        const int    kwbase = c * KC;
#pragma unroll
        for (int kk = 0; kk < KC; kk += 4) {
            v2f a = *(const v2f*)(tb + ((wv << 4) + m) * LSTRIDE + kk + khi);
            v2f bf;
            bf.x = wlds[kwbase + kk + khi];
            bf.y = wlds[kwbase + kk + khi + 1];
            // (neg_a, A, neg_b, B, c_mod, C, reuse_a, reuse_b)
            acc = __builtin_amdgcn_wmma_f32_16x16x4_f32(
                      false, a, false, bf, (short)0, acc, false, false);
        }
        __syncthreads();   // all waves done reading before buffer is rewritten
    }

    // Every column of D holds the row's partial dot; lane 0 owns rows 0-7
    // (VGPR 0-7), lane 16 owns rows 8-15. SCALE=2 applied per partial.
    const int rbase = row0 + (wv << 4);
    if (lane == 0) {
#pragma unroll
        for (int v = 0; v < 8; ++v)
            atomicAdd(&out[rbase + v], 2.0f * acc[v]);
    } else if (lane == 16) {
#pragma unroll
        for (int v = 0; v < 8; ++v)
            atomicAdd(&out[rbase + 8 + v], 2.0f * acc[v]);
    }
}

// ---------------------------------------------------------------------------
extern "C" void kernel_launch(void* const* d_in, const int* in_sizes, int n_in,
                              void* d_out, int out_size, void* d_ws, size_t ws_size,
                              hipStream_t stream) {
    const float* x = (const float*)d_in[0];   // [16384, 4096] f32
    const float* w = (const float*)d_in[1];   // [4096, 4096]  f32
    float* out  = (float*)d_out;              // [16384] f32
    float* wsum = (float*)d_ws;               // 4096 f32 scratch

    // Stage 0: zero out + wsum (deterministic every call).
    mn_zero_kernel<<<(BATCH + 255) / 256, 256, 0, stream>>>(out, wsum);

    // Stage 1: column sums of W.
    mn_colsum_kernel<<<256, 256, 0, stream>>>(w, wsum);

    // Stage 2: async-staged WMMA GEMV.
    // 256 row-groups x 4 K-chunks = 1024 blocks of 128 threads (4096 waves).
    mn_gemv_wmma_kernel<<<256 * KCHUNKS, THREADS, 0, stream>>>(x, wsum, out);
}